// SOMLayer_89111981458031
// MI455X (gfx1250) — compile-verified
//
#include <hip/hip_runtime.h>
#include <hip/hip_bf16.h>

typedef float v2f __attribute__((ext_vector_type(2)));
typedef float v8f __attribute__((ext_vector_type(8)));
typedef int   v4i __attribute__((vector_size(16)));

#define N_INPUTS   16384
#define DIMS       256
#define N_PROTO    1024
#define SQRT3_2    0.86602540378443864676f
#define LDS_STRIDE 260   // 256 + 4 pad: row stride 1040B (16B aligned), 4-bank rotation kills conflicts

// ---------------------------------------------------------------------------
// gfx1250 async global->LDS (ASYNCcnt path), with safe fallback
// Probe-confirmed prototype: (v4i __device__*, v4i __shared__*, imm offset, imm cpol)
// ---------------------------------------------------------------------------
#if __has_builtin(__builtin_amdgcn_global_load_async_to_lds_b128)
#define HAVE_ASYNC_LDS 1
typedef __attribute__((address_space(1))) v4i* gl_v4i_p;
typedef __attribute__((address_space(3))) v4i* lds_v4i_p;
#endif

__device__ __forceinline__ void async_ld_b128(void* lds_dst, const void* gsrc) {
#ifdef HAVE_ASYNC_LDS
  __builtin_amdgcn_global_load_async_to_lds_b128(
      (gl_v4i_p)gsrc, (lds_v4i_p)lds_dst, 0, 0);
#else
  *(float4*)lds_dst = *(const float4*)gsrc;
#endif
}

__device__ __forceinline__ void wait_async_lds() {
#ifdef HAVE_ASYNC_LDS
#if __has_builtin(__builtin_amdgcn_s_wait_asynccnt)
  __builtin_amdgcn_s_wait_asynccnt(0);
#else
  asm volatile("s_wait_asynccnt 0" ::: "memory");
#endif
#endif
}

// ---------------------------------------------------------------------------
// Kernel 1: build the hexagonal-PBC Gaussian neighborhood H[1024][1024]
// ---------------------------------------------------------------------------
__global__ __launch_bounds__(256) void build_H(float* __restrict__ H) {
  const int idx = blockIdx.x * 256 + threadIdx.x;       // 0 .. 1024*1024-1
  const int a = idx >> 10, b = idx & 1023;
  const int ra = a >> 5, ca = a & 31;
  const int rb = b >> 5, cb = b & 31;
  const float xa = (float)ca + 0.5f * (float)(ra & 1);
  const float ya = (float)ra * SQRT3_2;
  const float xb = (float)cb + 0.5f * (float)(rb & 1);
  const float yb = (float)rb * SQRT3_2;
  const float dx0 = xa - xb, dy0 = ya - yb;
  const float W = 32.0f, Hh = 32.0f * SQRT3_2;
  float d2 = 3.0e38f;
  #pragma unroll
  for (int sx = -1; sx <= 1; ++sx)
    #pragma unroll
    for (int sy = -1; sy <= 1; ++sy) {
      const float dx = dx0 + (float)sx * W;
      const float dy = dy0 + (float)sy * Hh;
      d2 = fminf(d2, fmaf(dx, dx, dy * dy));
    }
  H[idx] = expf(-0.5f * d2);                            // sigma = 1 -> /(2*sigma^2)
}

// ---------------------------------------------------------------------------
// Kernel 2: per-row sum of squares (one wave32 per row).  rows = gridDim.x*8
// ---------------------------------------------------------------------------
__global__ __launch_bounds__(256) void rowsq(const float* __restrict__ A,
                                             float* __restrict__ out) {
  const int wave = threadIdx.x >> 5, lane = threadIdx.x & 31;
  const int row = blockIdx.x * 8 + wave;
  const float* p = A + (size_t)row * DIMS;
  float s = 0.0f;
  #pragma unroll
  for (int c = lane; c < DIMS; c += 32) { const float v = p[c]; s = fmaf(v, v, s); }
  #pragma unroll
  for (int off = 16; off; off >>= 1) s += __shfl_xor(s, off, 32);
  if (lane == 0) out[row] = s;
}

// ---------------------------------------------------------------------------
// Kernel 3: hsum[m] = sum_j H[m][j];  q[m] = sum_j H[m][j]*p2[j]
// ---------------------------------------------------------------------------
__global__ __launch_bounds__(256) void hsum_q(const float* __restrict__ H,
                                              const float* __restrict__ p2,
                                              float* __restrict__ hsum,
                                              float* __restrict__ q) {
  __shared__ float s1[8], s2[8];
  const int m = blockIdx.x, t = threadIdx.x;
  const float* hr = H + (size_t)m * N_PROTO;
  float a = 0.0f, b = 0.0f;
  for (int j = t; j < N_PROTO; j += 256) {
    const float h = hr[j];
    a += h;
    b = fmaf(h, p2[j], b);
  }
  #pragma unroll
  for (int off = 16; off; off >>= 1) { a += __shfl_xor(a, off, 32); b += __shfl_xor(b, off, 32); }
  const int wave = t >> 5, lane = t & 31;
  if (lane == 0) { s1[wave] = a; s2[wave] = b; }
  __syncthreads();
  if (t == 0) {
    float A = 0.0f, B = 0.0f;
    #pragma unroll
    for (int w = 0; w < 8; ++w) { A += s1[w]; B += s2[w]; }
    hsum[m] = A; q[m] = B;
  }
}

// ---------------------------------------------------------------------------
// Kernel 4 (WMMA): G[d][m] = sum_j P[j][d] * H[j][m]
// grid (16 d-tiles, 8 m-slabs); WG stages P[:, d0:d0+16] (64 KB) in LDS via
// async loads; each wave owns one 16x16 output tile, K=1024 in steps of 4.
// A-operand: transposed-P from LDS (row stride 16 floats -> conflict-free).
// ---------------------------------------------------------------------------
__global__ __launch_bounds__(256) void compG_wmma(const float* __restrict__ P,
                                                  const float* __restrict__ H,
                                                  float* __restrict__ G) {
  extern __shared__ float Pl[];                 // [1024][16] = 64 KB
  const int tid = threadIdx.x;
  const int d0 = blockIdx.x * 16;
  const int m0 = blockIdx.y * 128;

  // stage P column block: 4096 x b128 async transfers
  for (int i = tid; i < 4096; i += 256) {
    const int j = i >> 2, c4 = (i & 3) << 2;
    async_ld_b128(&Pl[j * 16 + c4], P + (size_t)j * DIMS + d0 + c4);
  }
  wait_async_lds();
  __syncthreads();

  const int wave = tid >> 5, lane = tid & 31;
  const int half = lane >> 4, mlane = lane & 15;
  const int mcol = m0 + wave * 16 + mlane;

  const float* aP = &Pl[2 * half * 16 + mlane];                   // A[m=mlane][k pair]
  const float* bH = H + (size_t)(2 * half) * N_PROTO + mcol;      // B[k pair][n=mcol]

  v8f acc = {};
  #pragma unroll 8
  for (int j = 0; j < N_PROTO; j += 4) {
    v2f a, b;
    a.x = aP[j * 16];                    // P[j + 2*half    ][d0+mlane]
    a.y = aP[j * 16 + 16];               // P[j + 2*half + 1][d0+mlane]
    b.x = bH[(size_t)j * N_PROTO];            // H[j + 2*half    ][mcol]
    b.y = bH[(size_t)j * N_PROTO + N_PROTO];  // H[j + 2*half + 1][mcol]
    acc = __builtin_amdgcn_wmma_f32_16x16x4_f32(false, a, false, b,
                                                (short)0, acc, false, false);
  }
  #pragma unroll
  for (int r = 0; r < 8; ++r)
    G[(size_t)(d0 + r + 8 * half) * N_PROTO + mcol] = acc[r];
}

// ---------------------------------------------------------------------------
// Kernel 5: fused  E = x2*hsum - 2*(X@G) + q  ->  out = 0.5*min_m E
// 256 workgroups x 64 rows; 8 waves each own a 128-col slice (8 x 16-col tiles).
// fp32 WMMA 16x16x4 (exact fp32 GEMM on the matrix pipes).
// ---------------------------------------------------------------------------
__global__ __launch_bounds__(256) void som_gemm(const float* __restrict__ X,
                                                const float* __restrict__ G,
                                                const float* __restrict__ hsum,
                                                const float* __restrict__ q,
                                                const float* __restrict__ x2,
                                                float* __restrict__ out) {
  extern __shared__ float smem[];
  float* Xs      = smem;                         // 64 * LDS_STRIDE
  float* x2s     = smem + 64 * LDS_STRIDE;       // 64
  float* partial = x2s + 64;                     // 8 * 64

  const int tid = threadIdx.x;
  const int n0  = blockIdx.x * 64;

  // async stage of the 64x256 X tile into padded LDS rows
  for (int i = tid; i < 64 * 64; i += 256) {
    const int row = i >> 6, c4 = (i & 63) << 2;
    async_ld_b128(&Xs[row * LDS_STRIDE + c4], X + (size_t)(n0 + row) * DIMS + c4);
  }
  if (tid < 64) x2s[tid] = x2[n0 + tid];
  wait_async_lds();
  __syncthreads();

  const int wave  = tid >> 5, lane = tid & 31;
  const int half  = lane >> 4, mlane = lane & 15;
  const int cbase = wave * 128;

  // A operand base: lane 'mlane' = matrix row, half selects K pair {0,1}/{2,3}
  const float* a0 = &Xs[mlane * LDS_STRIDE + 2 * half];

  v8f runmin[4];
  #pragma unroll
  for (int t = 0; t < 4; ++t)
    #pragma unroll
    for (int r = 0; r < 8; ++r) runmin[t][r] = 3.0e38f;

  for (int ct = 0; ct < 8; ++ct) {
    const int c0 = cbase + ct * 16;
    // prefetch next column tile of G into caches
    if (ct < 7) __builtin_prefetch(&G[(size_t)(2 * half) * N_PROTO + c0 + 16 + mlane], 0, 1);

    v8f acc[4] = {};
    const float* gB = G + (size_t)(2 * half) * N_PROTO + c0 + mlane;  // B lane = n + 16*(k/2)
    #pragma unroll 4
    for (int k = 0; k < DIMS; k += 4) {
      v2f b;
      b.x = gB[(size_t)k * N_PROTO];            // B[k + 2*half    ][n]
      b.y = gB[(size_t)k * N_PROTO + N_PROTO];  // B[k + 2*half + 1][n]
      #pragma unroll
      for (int t = 0; t < 4; ++t) {
        const v2f a = *(const v2f*)(a0 + t * 16 * LDS_STRIDE + k);
        acc[t] = __builtin_amdgcn_wmma_f32_16x16x4_f32(
            /*neg_a=*/false, a, /*neg_b=*/false, b,
            /*c_mod=*/(short)0, acc[t], /*reuse_a=*/false, /*reuse_b=*/false);
      }
    }
    // epilogue for this 16-col tile: E = x2*hsum + q - 2*acc, fold into running min
    const float hv = hsum[c0 + mlane];
    const float qv = q[c0 + mlane];
    #pragma unroll
    for (int t = 0; t < 4; ++t)
      #pragma unroll
      for (int r = 0; r < 8; ++r) {
        const float x2v = x2s[16 * t + r + 8 * half];   // LDS broadcast read
        const float e = fmaf(x2v, hv, qv) - 2.0f * acc[t][r];
        runmin[t][r] = fminf(runmin[t][r], e);
      }
  }

  __syncthreads();  // all waves done reading Xs; partial[] lives after it

  // min across the 16 column-lanes of each half, then publish per-row mins
  #pragma unroll
  for (int t = 0; t < 4; ++t)
    #pragma unroll
    for (int r = 0; r < 8; ++r) {
      float v = runmin[t][r];
      v = fminf(v, __shfl_xor(v, 1, 32));
      v = fminf(v, __shfl_xor(v, 2, 32));
      v = fminf(v, __shfl_xor(v, 4, 32));
      v = fminf(v, __shfl_xor(v, 8, 32));
      if (mlane == 0) partial[wave * 64 + 16 * t + r + 8 * half] = v;
    }
  __syncthreads();

  if (tid < 64) {
    float m = partial[tid];
    #pragma unroll
    for (int w = 1; w < 8; ++w) m = fminf(m, partial[w * 64 + tid]);
    out[n0 + tid] = 0.5f * m;
  }
}

// ---------------------------------------------------------------------------
extern "C" void kernel_launch(void* const* d_in, const int* in_sizes, int n_in,
                              void* d_out, int out_size, void* d_ws, size_t ws_size,
                              hipStream_t stream) {
  const float* X = (const float*)d_in[0];   // [16384, 256]
  const float* P = (const float*)d_in[1];   // [1024, 256]
  float* out = (float*)d_out;               // [1, 16384]

  float* ws = (float*)d_ws;                 // ~5.32 MB total
  float* H   = ws;                          // 1024*1024
  float* G   = H  + N_PROTO * N_PROTO;      // 256*1024
  float* p2  = G  + DIMS * N_PROTO;         // 1024
  float* hs  = p2 + N_PROTO;                // 1024
  float* qv  = hs + N_PROTO;                // 1024
  float* x2  = qv + N_PROTO;                // 16384

  build_H<<<(N_PROTO * N_PROTO) / 256, 256, 0, stream>>>(H);
  rowsq<<<N_PROTO / 8, 256, 0, stream>>>(P, p2);
  rowsq<<<N_INPUTS / 8, 256, 0, stream>>>(X, x2);
  hsum_q<<<N_PROTO, 256, 0, stream>>>(H, p2, hs, qv);
  compG_wmma<<<dim3(DIMS / 16, N_PROTO / 128), 256, 1024 * 16 * sizeof(float), stream>>>(P, H, G);

  const size_t smem_bytes = (64 * LDS_STRIDE + 64 + 8 * 64) * sizeof(float);
  som_gemm<<<N_INPUTS / 64, 256, smem_bytes, stream>>>(X, G, hs, qv, x2, out);
}